// CRF_Layer_21388937134526
// MI455X (gfx1250) — compile-verified
//
#include <hip/hip_runtime.h>

#define Nw 2048
#define Mm 64
#define Kk 32
#define Dd 512
#define NCOPY 8
#define ACC_STRIDE (Kk*Dd + Kk*Kk)   // 16384 + 1024 = 17408

typedef float v2f __attribute__((ext_vector_type(2)));
typedef float v8f __attribute__((ext_vector_type(8)));

// D = A(16x4 f32) * B(4x16 f32) + C(16x16 f32)
__device__ __forceinline__ v8f wmma4(v2f a, v2f b, v8f c) {
  return __builtin_amdgcn_wmma_f32_16x16x4_f32(false, a, false, b, (short)0, c, false, false);
}

__device__ __forceinline__ float wave_max(float v) {
  #pragma unroll
  for (int off = 16; off > 0; off >>= 1) v = fmaxf(v, __shfl_xor(v, off, 32));
  return v;
}
__device__ __forceinline__ float wave_sum(float v) {
  #pragma unroll
  for (int off = 16; off > 0; off >>= 1) v += __shfl_xor(v, off, 32);
  return v;
}

__global__ __launch_bounds__(256)
void crf_word_kernel(const float* __restrict__ Wg, const float* __restrict__ Tg,
                     const float* __restrict__ data, const int* __restrict__ labels,
                     float* __restrict__ acc)
{
  __shared__ float sDots [Mm][Kk + 1];
  __shared__ float sAlpha[Mm][Kk + 1];
  __shared__ float sBeta [Mm][Kk + 1];
  __shared__ float sU    [Mm][Kk + 1];
  __shared__ float sV    [Mm][Kk + 1];
  __shared__ float sG    [Mm][Kk + 1];
  __shared__ float sT    [Kk][Kk + 1];
  __shared__ float sCnt  [Kk][Kk + 1];
  __shared__ float sCu[Mm];
  __shared__ float sDv[Mm];
  __shared__ int   sLab[Mm];
  __shared__ float sLogZ;

  const int tid  = threadIdx.x;
  const int w    = tid >> 5;
  const int lane = tid & 31;
  const int col  = lane & 15;
  const int hi   = lane >> 4;
  const int word = blockIdx.x;
  const float* dptr = data + (size_t)word * (Mm * Dd);

  float* base  = acc + (size_t)(blockIdx.x & (NCOPY - 1)) * ACC_STRIDE;
  float* accDW = base;
  float* accDT = base + Kk * Dd;

  // Phase 0: stage T, zero transition-count tile, stage labels
  for (int i = tid; i < Kk * Kk; i += 256) {
    sT  [i >> 5][i & 31] = Tg[i];
    sCnt[i >> 5][i & 31] = 0.f;
  }
  if (tid < Mm) sLab[tid] = labels[word * Mm + tid];

  // ---- Phase 1: dots = data @ W^T via f32 WMMA; one 16x16 tile per wave ----
  // Two accumulator chains to hide WMMA latency.
  {
    const int m0 = (w >> 1) * 16;
    const int n0 = (w & 1) * 16;
    v8f c0 = {}, c1 = {};
    const float* arow = dptr + (size_t)(m0 + col) * Dd + 2 * hi;
    const float* brow = Wg   + (size_t)(n0 + col) * Dd + 2 * hi;
    #pragma unroll 4
    for (int kk = 0; kk < Dd; kk += 8) {
      v2f a0 = *(const v2f*)(arow + kk);       // A: M=col, K=kk+2*hi, kk+2*hi+1
      v2f b0 = *(const v2f*)(brow + kk);       // B: N=col, K=kk+2*hi, kk+2*hi+1
      v2f a1 = *(const v2f*)(arow + kk + 4);
      v2f b1 = *(const v2f*)(brow + kk + 4);
      c0 = wmma4(a0, b0, c0);
      c1 = wmma4(a1, b1, c1);
    }
    v8f c = c0 + c1;
    #pragma unroll
    for (int r = 0; r < 8; ++r)
      sDots[m0 + r + 8 * hi][n0 + col] = c[r];
  }
  __syncthreads();

  // ---- Phase 2: alpha (wave 0) and beta (wave 1) recursions, lane = state k;
  //      idle waves 2..7 prefetch the data tile for phase 6 ----
  if (w == 0) {
    float Tc[32];
    #pragma unroll
    for (int j = 0; j < 32; ++j) Tc[j] = sT[j][lane];   // T[j][k], k = lane
    float prev = 0.f;
    sAlpha[0][lane] = 0.f;
    for (int i = 1; i < Mm; ++i) {
      float sv = prev + sDots[i - 1][lane];
      float tv[32];
      float mx = -3.0e38f;
      #pragma unroll
      for (int j = 0; j < 32; ++j) {
        float s = __shfl(sv, j, 32);
        tv[j] = s + Tc[j];
        mx = fmaxf(mx, tv[j]);
      }
      float sum = 0.f;
      #pragma unroll
      for (int j = 0; j < 32; ++j) sum += __expf(tv[j] - mx);
      prev = mx + __logf(sum);
      sAlpha[i][lane] = prev;
    }
  } else if (w == 1) {
    float Tr[32];
    #pragma unroll
    for (int j = 0; j < 32; ++j) Tr[j] = sT[lane][j];   // T[k][j], k = lane
    float prev = 0.f;
    sBeta[Mm - 1][lane] = 0.f;
    for (int i = Mm - 2; i >= 0; --i) {
      float sv = prev + sDots[i + 1][lane];
      float tv[32];
      float mx = -3.0e38f;
      #pragma unroll
      for (int j = 0; j < 32; ++j) {
        float s = __shfl(sv, j, 32);
        tv[j] = s + Tr[j];
        mx = fmaxf(mx, tv[j]);
      }
      float sum = 0.f;
      #pragma unroll
      for (int j = 0; j < 32; ++j) sum += __expf(tv[j] - mx);
      prev = mx + __logf(sum);
      sBeta[i][lane] = prev;
    }
  } else {
    // warm near caches for the phase-6 re-read of the 128 KB data tile
    // (one global_prefetch_b8 per 128B line; no counters, overlaps recursion)
    const int t2 = tid - 64;                  // 0..191
    for (int line = t2; line < (Mm * Dd) / 32; line += 192)
      __builtin_prefetch(dptr + line * 32, 0, 3);
  }
  __syncthreads();

  // ---- Phase 3: logZ = lse(alpha[M-1] + dots[M-1]) ----
  if (w == 0) {
    float v  = sAlpha[Mm - 1][lane] + sDots[Mm - 1][lane];
    float mx = wave_max(v);
    float s  = wave_sum(__expf(v - mx));
    if (lane == 0) sLogZ = mx + __logf(s);
  }
  __syncthreads();
  const float logZ = sLogZ;

  // ---- Phase 4: p1 -> G = onehot - p1; U/V factors; transition counts ----
  #pragma unroll 1
  for (int t = 0; t < 8; ++t) {
    const int i = w + 8 * t;
    const float a = sAlpha[i][lane], b = sBeta[i][lane], d = sDots[i][lane];
    float e   = a + b + d;
    float mx  = wave_max(e);
    float p   = __expf(e - mx);
    float inv = 1.f / wave_sum(p);
    sG[i][lane] = ((lane == sLab[i]) ? 1.f : 0.f) - p * inv;

    float aL = a + d;
    float ca = wave_max(aL);
    if (i < Mm - 1) { sU[i][lane] = __expf(aL - ca); if (lane == 0) sCu[i] = ca; }
    else            { sU[Mm - 1][lane] = 0.f; }

    float bR = b + d;
    float cb = wave_max(bR);
    if (i >= 1) { sV[i - 1][lane] = __expf(bR - cb); if (lane == 0) sDv[i - 1] = cb; }
    else        { sV[Mm - 1][lane] = 0.f; }
  }
  if (tid < Mm - 1) atomicAdd(&sCnt[sLab[tid]][sLab[tid + 1]], 1.f);
  __syncthreads();

  // fold per-position pair weight exp(c_i + d_{i+1} - logZ) into U'
  for (int idx = tid; idx < (Mm - 1) * Kk; idx += 256) {
    const int i = idx >> 5, k = idx & 31;
    sU[i][k] *= __expf(sCu[i] + sDv[i] - logZ);
  }
  __syncthreads();

  // ---- Phase 5: sumP2 = exp(T) .* (U'^T V) via WMMA; dT contribution ----
  if (w < 4) {
    const int j0 = (w >> 1) * 16, k0 = (w & 1) * 16;
    v8f c0 = {}, c1 = {};
    #pragma unroll
    for (int ii = 0; ii < Mm; ii += 8) {
      v2f a0, b0, a1, b1;
      a0.x = sU[ii + 2 * hi    ][j0 + col];
      a0.y = sU[ii + 2 * hi + 1][j0 + col];
      b0.x = sV[ii + 2 * hi    ][k0 + col];
      b0.y = sV[ii + 2 * hi + 1][k0 + col];
      a1.x = sU[ii + 4 + 2 * hi    ][j0 + col];
      a1.y = sU[ii + 4 + 2 * hi + 1][j0 + col];
      b1.x = sV[ii + 4 + 2 * hi    ][k0 + col];
      b1.y = sV[ii + 4 + 2 * hi + 1][k0 + col];
      c0 = wmma4(a0, b0, c0);
      c1 = wmma4(a1, b1, c1);
    }
    v8f c = c0 + c1;
    #pragma unroll
    for (int r = 0; r < 8; ++r) {
      const int j = j0 + r + 8 * hi, k = k0 + col;
      const float val = sCnt[j][k] - __expf(sT[j][k]) * c[r];
      atomicAdd(accDT + j * Kk + k, val);
    }
  }

  // ---- Phase 6: dw = G^T @ data via WMMA; 8 tiles per wave ----
  // Hoist A-fragments (invariant over d-tiles) into registers once.
  {
    const int j0 = (w & 1) * 16;
    v2f afr[16];
    #pragma unroll
    for (int s = 0; s < 16; ++s) {
      afr[s].x = sG[4 * s + 2 * hi    ][j0 + col];
      afr[s].y = sG[4 * s + 2 * hi + 1][j0 + col];
    }
    #pragma unroll 1
    for (int dt = (w >> 1); dt < 32; dt += 4) {
      const int d0 = dt * 16;
      v8f c0 = {}, c1 = {};
      #pragma unroll
      for (int s = 0; s < 16; s += 2) {
        v2f b0, b1;
        b0.x = dptr[(size_t)(4 * s + 2 * hi    ) * Dd + d0 + col];
        b0.y = dptr[(size_t)(4 * s + 2 * hi + 1) * Dd + d0 + col];
        b1.x = dptr[(size_t)(4 * s + 4 + 2 * hi    ) * Dd + d0 + col];
        b1.y = dptr[(size_t)(4 * s + 4 + 2 * hi + 1) * Dd + d0 + col];
        c0 = wmma4(afr[s],     b0, c0);
        c1 = wmma4(afr[s + 1], b1, c1);
      }
      v8f c = c0 + c1;
      #pragma unroll
      for (int r = 0; r < 8; ++r)
        atomicAdd(accDW + (j0 + r + 8 * hi) * Dd + d0 + col, c[r]);
    }
  }
}

__global__ void zero_ws_kernel(float* __restrict__ p, int n) {
  const int i = blockIdx.x * 256 + threadIdx.x;
  if (i < n) p[i] = 0.f;
}

__global__ void reduce_out_kernel(const float* __restrict__ acc, float* __restrict__ out) {
  const int i = blockIdx.x * 256 + threadIdx.x;
  if (i >= ACC_STRIDE) return;
  float s = 0.f;
  #pragma unroll
  for (int c = 0; c < NCOPY; ++c) s += acc[(size_t)c * ACC_STRIDE + i];
  out[i] = s * (1.0f / (float)Nw);   // mean over words
}

extern "C" void kernel_launch(void* const* d_in, const int* in_sizes, int n_in,
                              void* d_out, int out_size, void* d_ws, size_t ws_size,
                              hipStream_t stream) {
  const float* W      = (const float*)d_in[0];   // [32,512]
  const float* T      = (const float*)d_in[1];   // [32,32]
  const float* data   = (const float*)d_in[2];   // [2048,64,512]
  const int*   labels = (const int*)  d_in[3];   // [2048,64]
  float* out = (float*)d_out;                    // 17408 f32
  float* ws  = (float*)d_ws;                     // NCOPY * 17408 f32 accumulators

  const int nacc = NCOPY * ACC_STRIDE;
  zero_ws_kernel<<<(nacc + 255) / 256, 256, 0, stream>>>(ws, nacc);
  crf_word_kernel<<<Nw, 256, 0, stream>>>(W, T, data, labels, ws);
  reduce_out_kernel<<<(ACC_STRIDE + 255) / 256, 256, 0, stream>>>(ws, out);
}